// Pytorch3dMeshRenderer_43087111914136
// MI455X (gfx1250) — compile-verified
//
#include <hip/hip_runtime.h>

typedef __attribute__((ext_vector_type(2))) float v2f;
typedef __attribute__((ext_vector_type(8))) float v8f;
typedef int v4i_ __attribute__((vector_size(16)));
typedef __attribute__((address_space(1))) v4i_* gv4i_ptr;   // global int4*
typedef __attribute__((address_space(3))) v4i_* lv4i_ptr;   // LDS int4*

#define IMG 256
#define BIGZ 1.0e10f
#define WAVES_PER_BLOCK 8
#define STAGE_FACES 64                    // faces staged to LDS per step
#define CHUNKS_PER_STAGE (STAGE_FACES / 16)
#define STAGE_F4 (STAGE_FACES * 3)        // 192 float4 per stage (3KB)

#if __has_builtin(__builtin_amdgcn_global_load_async_to_lds_b128) && \
    __has_builtin(__builtin_amdgcn_s_wait_asynccnt)
#define USE_ASYNC_LDS 1
#else
#define USE_ASYNC_LDS 0
#endif

// ---------------------------------------------------------------------------
// Pass 1: per-face affine coefficients for w0, w1, z as functions of (px,py).
//   w0 = a0*px + b0*py + g0 ; w1 = a1*px + b1*py + g1 ; z = za*px + zb*py + zg
// The .w slot of every float4 is 0 so (g, 0) can be loaded as one b64.
// Degenerate/padded faces get g0 = -1e30 so the inside test always fails.
// ---------------------------------------------------------------------------
__global__ void prep_kernel(const float* __restrict__ verts,
                            const int*   __restrict__ faces,
                            float4*      __restrict__ coef,
                            int F, int Fpad) {
    int f = blockIdx.x * blockDim.x + threadIdx.x;
    if (f >= Fpad) return;
    float4 c0, c1, c2;
    if (f < F) {
        int ia = faces[3*f+0], ib = faces[3*f+1], ic = faces[3*f+2];
        float ax = verts[3*ia+0], ay = verts[3*ia+1], az = verts[3*ia+2] + 10.0f;
        float bx = verts[3*ib+0], by = verts[3*ib+1], bz = verts[3*ib+2] + 10.0f;
        float cx = verts[3*ic+0], cy = verts[3*ic+1], cz = verts[3*ic+2] + 10.0f;
        float denom = (by - cy) * (ax - cx) + (cx - bx) * (ay - cy);
        bool  valid = fabsf(denom) > 1e-8f;
        float inv   = valid ? (1.0f / denom) : 0.0f;
        float e0x = by - cy, e0y = cx - bx;
        float e1x = cy - ay, e1y = ax - cx;
        float a0 = e0x * inv, b0 = e0y * inv, g0 = -(e0x * cx + e0y * cy) * inv;
        float a1 = e1x * inv, b1 = e1y * inv, g1 = -(e1x * cx + e1y * cy) * inv;
        float azc = az - cz, bzc = bz - cz;
        float za = a0 * azc + a1 * bzc;
        float zb = b0 * azc + b1 * bzc;
        float zg = cz + g0 * azc + g1 * bzc;
        if (!valid) { a0 = 0.0f; b0 = 0.0f; g0 = -1.0e30f; a1 = 0.0f; b1 = 0.0f; g1 = 0.0f; }
        c0 = make_float4(a0, b0, g0, 0.0f);
        c1 = make_float4(a1, b1, g1, 0.0f);
        c2 = make_float4(za, zb, zg, 0.0f);
    } else {
        c0 = make_float4(0.0f, 0.0f, -1.0e30f, 0.0f);
        c1 = make_float4(0.0f, 0.0f, 0.0f, 0.0f);
        c2 = make_float4(0.0f, 0.0f, BIGZ, 0.0f);
    }
    coef[3*f+0] = c0; coef[3*f+1] = c1; coef[3*f+2] = c2;
}

// ---------------------------------------------------------------------------
// Pass 2: WMMA rasterization with double-buffered async LDS staging.
// Each wave32 owns a 16-pixel strip; 8 waves per block share the face stream.
//   A (16x4): row m = [px, py, 1, 0]; lanes 0-15 -> K0,K1; lanes 16-31 -> K2,K3
//   B (4x16): col n = face n coeffs (same K split) -> one ds_load_b64 per lane
//   D (16x16): VGPR v, lane L -> pixel m = v + 8*(L>=16), face n = L&15
// ---------------------------------------------------------------------------
__global__ void __launch_bounds__(256) raster_kernel(
        const float4* __restrict__ coef,
        const int*    __restrict__ faces,
        const float*  __restrict__ attrs,
        float*        __restrict__ out,
        int nstages) {
    __shared__ float4 cbuf[2][STAGE_F4];                 // 6KB double buffer
    __shared__ float  lz[WAVES_PER_BLOCK][16][16];
    __shared__ int    lf[WAVES_PER_BLOCK][16][16];

    const int tid    = threadIdx.x;
    const int w      = tid >> 5;
    const int lane   = tid & 31;
    const int lane16 = lane & 15;
    const bool hi    = lane >= 16;

    const int strip   = blockIdx.x * WAVES_PER_BLOCK + w;
    const int y       = strip >> 4;
    const int xs_base = (strip & 15) << 4;

    const float py = ((float)y + 0.5f) * (1.0f / 128.0f) - 1.0f;

    // A-matrix fragment (constant across all chunks).
    v2f amat;
    {
        float pxm = ((float)(xs_base + lane16) + 0.5f) * (1.0f / 128.0f) - 1.0f;
        if (!hi) { amat.x = pxm;  amat.y = py;   }   // K0 = px, K1 = py
        else     { amat.x = 1.0f; amat.y = 0.0f; }   // K2 = 1,  K3 = 0
    }

    float zbest[8];
    int   fbest[8];
#pragma unroll
    for (int v = 0; v < 8; ++v) { zbest[v] = BIGZ; fbest[v] = 0x7FFFFFFF; }

    const v8f czero = {};

    // ---- staging: global -> LDS (async if available), 192 b128 per stage ----
    auto issue_stage = [&](int s) {
        if (tid < STAGE_F4) {                          // waves 0..5, uniform per wave
            const float4* src = coef + s * STAGE_F4 + tid;
            float4*       dst = &cbuf[s & 1][tid];
#if USE_ASYNC_LDS
            __builtin_amdgcn_global_load_async_to_lds_b128(
                (gv4i_ptr)(unsigned long long)src,
                (lv4i_ptr)(unsigned)(unsigned long long)dst,
                0, 0);
#else
            *dst = *src;
#endif
        }
    };

    // ---- consume one 64-face stage from LDS: 4 chunks x (3 ds_b64 + 3 wmma) ----
    auto process_stage = [&](int s) {
        const int nb     = s & 1;
        const int fbase0 = s * STAGE_FACES;
#pragma unroll
        for (int c = 0; c < CHUNKS_PER_STAGE; ++c) {
            const int fl = (c << 4) + lane16;          // local face 0..63
            const char* p = (const char*)(&cbuf[nb][fl * 3]) + (hi ? 8 : 0);
            v2f b0 = *(const v2f*)(p + 0);             // lo:(a0,b0)  hi:(g0,0)
            v2f b1 = *(const v2f*)(p + 16);            // lo:(a1,b1)  hi:(g1,0)
            v2f b2 = *(const v2f*)(p + 32);            // lo:(za,zb)  hi:(zg,0)

            v8f w0v = __builtin_amdgcn_wmma_f32_16x16x4_f32(false, amat, false, b0, (short)0, czero, false, false);
            v8f w1v = __builtin_amdgcn_wmma_f32_16x16x4_f32(false, amat, false, b1, (short)0, czero, false, false);
            v8f zv  = __builtin_amdgcn_wmma_f32_16x16x4_f32(false, amat, false, b2, (short)0, czero, false, false);

            const int fj = fbase0 + fl;
#pragma unroll
            for (int v = 0; v < 8; ++v) {
                float w0 = w0v[v], w1 = w1v[v];
                float w2 = 1.0f - w0 - w1;
                float mn = fminf(fminf(w0, w1), w2);
                float z  = zv[v];
                bool take = (mn >= 0.0f) && (z < zbest[v]);
                if (take) { zbest[v] = z; fbest[v] = fj; }
            }
        }
    };

    // ---- software pipeline: stage s+1 in flight while s is consumed ----
    issue_stage(0);
    int s = 0;
    for (; s < nstages - 1; ++s) {
        issue_stage(s + 1);
#if USE_ASYNC_LDS
        __builtin_amdgcn_s_wait_asynccnt(1);           // stage s done, s+1 in flight
#endif
        __syncthreads();                               // LDS visible block-wide
        process_stage(s);
        __syncthreads();                               // done reading buf[s&1]
    }
#if USE_ASYNC_LDS
    __builtin_amdgcn_s_wait_asynccnt(0);
#endif
    __syncthreads();
    process_stage(s);

    // ---- per-pixel argmin reduction across the 16 face columns ----
#pragma unroll
    for (int v = 0; v < 8; ++v) {
        int m = hi ? (v + 8) : v;
        lz[w][m][lane16] = zbest[v];
        lf[w][m][lane16] = fbest[v];
    }
    __syncthreads();

    if (lane < 16) {
        float bzv = BIGZ;
        int   bfv = 0x7FFFFFFF;
#pragma unroll
        for (int n = 0; n < 16; ++n) {
            float z = lz[w][lane][n];
            int   f = lf[w][lane][n];
            if (z < bzv || (z == bzv && f < bfv)) { bzv = z; bfv = f; }
        }
        const int x = xs_base + lane;
        float r = 0.0f, g = 0.0f, b = 0.0f, al = 0.0f;
        if (bzv < BIGZ) {
            float pxm = ((float)x + 0.5f) * (1.0f / 128.0f) - 1.0f;
            float4 c0 = coef[3*bfv+0];
            float4 c1 = coef[3*bfv+1];
            float w0 = fmaf(c0.x, pxm, fmaf(c0.y, py, c0.z));
            float w1 = fmaf(c1.x, pxm, fmaf(c1.y, py, c1.z));
            float w2 = 1.0f - w0 - w1;
            int ia = faces[3*bfv+0], ib = faces[3*bfv+1], ic = faces[3*bfv+2];
            r = w0 * attrs[3*ia+0] + w1 * attrs[3*ib+0] + w2 * attrs[3*ic+0];
            g = w0 * attrs[3*ia+1] + w1 * attrs[3*ib+1] + w2 * attrs[3*ic+1];
            b = w0 * attrs[3*ia+2] + w1 * attrs[3*ib+2] + w2 * attrs[3*ic+2];
            al = 1.0f;
        }
        const int pix = y * IMG + x;
        out[0 * IMG * IMG + pix] = r;
        out[1 * IMG * IMG + pix] = g;
        out[2 * IMG * IMG + pix] = b;
        out[3 * IMG * IMG + pix] = al;
    }
}

extern "C" void kernel_launch(void* const* d_in, const int* in_sizes, int n_in,
                              void* d_out, int out_size, void* d_ws, size_t ws_size,
                              hipStream_t stream) {
    const float* verts = (const float*)d_in[0];   // [1,V,3] f32
    const int*   faces = (const int*)d_in[1];     // [1,F,3] i32
    const float* attrs = (const float*)d_in[2];   // [1,V,3] f32
    float* out = (float*)d_out;                   // image(3*H*W) ++ alpha(H*W)

    const int F       = in_sizes[1] / 3;
    const int Fpad    = ((F + STAGE_FACES - 1) / STAGE_FACES) * STAGE_FACES;
    const int nstages = Fpad / STAGE_FACES;
    float4* coef = (float4*)d_ws;                 // Fpad * 3 * float4 (~468KB)

    prep_kernel<<<(Fpad + 255) / 256, 256, 0, stream>>>(verts, faces, coef, F, Fpad);

    const int strips = (IMG * IMG) / 16;          // 4096 strips
    const int blocks = strips / WAVES_PER_BLOCK;  // 512 blocks x 256 threads
    raster_kernel<<<blocks, 256, 0, stream>>>(coef, faces, attrs, out, nstages);
}